// GCNLayer_2774548873854
// MI455X (gfx1250) — compile-verified
//
#include <hip/hip_runtime.h>
#include <cstdint>

// COO SpMM: out[dst] += vals[e] * embeds[src], D_FEAT = 64, f32.
// Memory-bound (AI ~0.24 flop/byte). Whole working set (~40MB) fits MI455X's
// 192MB L2 -> optimize for L2-resident gather/atomic throughput + MLP.
// CDNA5 paths used: global_load_async_to_lds (ASYNCcnt double buffering),
// wave32 layout (32 lanes x float2 = one 64-wide row), hw global f32 atomics.

#define D_FEAT 64
#define WAVES_PER_BLOCK 8
#define THREADS (WAVES_PER_BLOCK * 32)
#define CHUNK 32
#define SCATTER_BLOCKS 512

typedef __attribute__((address_space(1))) float global_f32;

__device__ __forceinline__ void atomic_add_f32(float* p, float v) {
#if defined(__has_builtin)
#if __has_builtin(__builtin_amdgcn_global_atomic_fadd_f32)
  (void)__builtin_amdgcn_global_atomic_fadd_f32((global_f32*)(uintptr_t)p, v);
#else
  atomicAdd(p, v);
#endif
#else
  atomicAdd(p, v);
#endif
}

// Async global -> LDS copy, one b32 per active lane. lds_byte is the LDS byte
// address (low 32 bits of a generic pointer to a __shared__ object).
__device__ __forceinline__ void async_copy_b32(uint32_t lds_byte, const void* g) {
  asm volatile("global_load_async_to_lds_b32 %0, %1, off"
               :: "v"(lds_byte), "v"(g)
               : "memory");
}
__device__ __forceinline__ void wait_async_le0() { asm volatile("s_wait_asynccnt 0x0" ::: "memory"); }
__device__ __forceinline__ void wait_async_le3() { asm volatile("s_wait_asynccnt 0x3" ::: "memory"); }
__device__ __forceinline__ void wait_ds0()       { asm volatile("s_wait_dscnt 0x0"    ::: "memory"); }

__device__ __forceinline__ uint32_t lds_byte_off(const void* p) {
  // Generic pointers to LDS carry the LDS byte offset in their low 32 bits.
  return (uint32_t)(uintptr_t)p;
}

__global__ void __launch_bounds__(THREADS) spmm_scatter_kernel(
    const int* __restrict__ src_idx, const int* __restrict__ dst_idx,
    const float* __restrict__ vals, const float* __restrict__ embeds,
    float* __restrict__ out, int n_edges) {
  __shared__ int   s_src[WAVES_PER_BLOCK][2][CHUNK];
  __shared__ int   s_dst[WAVES_PER_BLOCK][2][CHUNK];
  __shared__ float s_val[WAVES_PER_BLOCK][2][CHUNK];

  const int lane = threadIdx.x & 31;
  const int wave = threadIdx.x >> 5;
  const int waveGlobal = blockIdx.x * WAVES_PER_BLOCK + wave;
  const int totalWaves = gridDim.x * WAVES_PER_BLOCK;
  const int nChunks = (n_edges + CHUNK - 1) / CHUNK;

  auto issue = [&](int chunk, int buf) {
    const int base = chunk * CHUNK;
    const int cnt = min(CHUNK, n_edges - base);
    wait_ds0();  // make sure prior DS reads of this buffer fully drained
    if (lane < cnt) {
      const uint32_t b = (uint32_t)(4 * lane);
      async_copy_b32(lds_byte_off(&s_src[wave][buf][0]) + b, src_idx + base + lane);
      async_copy_b32(lds_byte_off(&s_dst[wave][buf][0]) + b, dst_idx + base + lane);
      async_copy_b32(lds_byte_off(&s_val[wave][buf][0]) + b, vals    + base + lane);
    }
  };

  int c = waveGlobal;
  if (c >= nChunks) return;
  int p = 0;
  issue(c, p);  // prime buffer 0 (3 async ops outstanding)

  for (; c < nChunks; c += totalWaves) {
    const int cn = c + totalWaves;
    const bool hasNext = (cn < nChunks);
    if (hasNext) {
      issue(cn, p ^ 1);   // 6 outstanding
      wait_async_le3();   // current chunk's 3 ops (issued first, in-order) done
    } else {
      wait_async_le0();
    }

    const int base = c * CHUNK;
    const int cnt = min(CHUNK, n_edges - base);
#pragma unroll 4
    for (int j = 0; j < cnt; ++j) {
      const int s   = s_src[wave][p][j];   // wave-uniform broadcast reads
      const int d   = s_dst[wave][p][j];
      const float v = s_val[wave][p][j];
      // 32 lanes x float2 = one coalesced 256B row gather
      const float2 g =
          reinterpret_cast<const float2*>(embeds + (size_t)s * D_FEAT)[lane];
      float* o = out + (size_t)d * D_FEAT + 2 * lane;
      atomic_add_f32(o,     g.x * v);
      atomic_add_f32(o + 1, g.y * v);
    }
    p ^= 1;
  }
}

__global__ void zero_f32_kernel(float* __restrict__ out, int n) {
  int i = (blockIdx.x * blockDim.x + threadIdx.x) * 4;
  if (i + 3 < n) {
    *reinterpret_cast<float4*>(out + i) = make_float4(0.f, 0.f, 0.f, 0.f);
  } else {
    for (; i < n; ++i) out[i] = 0.f;
  }
}

extern "C" void kernel_launch(void* const* d_in, const int* in_sizes, int n_in,
                              void* d_out, int out_size, void* d_ws, size_t ws_size,
                              hipStream_t stream) {
  const int*   src    = (const int*)d_in[0];
  const int*   dst    = (const int*)d_in[1];
  const float* vals   = (const float*)d_in[2];
  const float* embeds = (const float*)d_in[3];
  float* out = (float*)d_out;
  const int n_edges = in_sizes[0];

  // 1) zero the (poisoned) output
  const int zThreads = 256;
  const int zPerBlock = zThreads * 4;
  const int zBlocks = (out_size + zPerBlock - 1) / zPerBlock;
  zero_f32_kernel<<<zBlocks, zThreads, 0, stream>>>(out, out_size);

  // 2) scatter-add: 512 blocks x 8 waves = 4096 waves, ~9 chunks each
  spmm_scatter_kernel<<<SCATTER_BLOCKS, THREADS, 0, stream>>>(
      src, dst, vals, embeds, out, n_edges);
}